// MechanicsModel_27479200759934
// MI455X (gfx1250) — compile-verified
//
#include <hip/hip_runtime.h>
#include <hip/hip_bf16.h>
#include <stdint.h>

// ---------------------------------------------------------------------------
// All-pairs Morse forces on MI455X (gfx1250).
//   e = exp(-(r - r0)/w), r0 = rad_i + rad_j, w=0.5, D=1.0
//   F_i = sum_j 4*(e*e - e)/r * (p_i - p_j)
//
// Transcendental-bound (v_rsq_f32 + v_exp_f32 per pair, 16.7M pairs);
// working set 64KB -> HBM irrelevant. CDNA5 path: async global->LDS tile
// staging (ASYNCcnt, double buffered) + broadcast ds_load_b128 reads.
//
// Packing trick: store w' = K*rad + 0.5 with K = log2(e)/w, so
//   E := 2e = exp2(pi.w' + pj.w' - K*r)   (one add + one fma + one v_exp)
//   mag = 4(e^2 - e) = E^2 - 2E
// Inactive particles get w' = -1e9 -> exp2 underflows to exactly 0 ->
// zero contribution and zero output row, no masking needed.
// ---------------------------------------------------------------------------

#define TILE   256   // threads per block == j-particles per LDS tile
#define JCHUNK 512   // j-range per block (2 tiles, double buffered)

// Builtin parameter types: (int4 AS1*, int4 AS3*, imm offset, imm cpol)
typedef int v4i __attribute__((vector_size(4 * sizeof(int))));
typedef __attribute__((address_space(1))) v4i* gptr_v4i;
typedef __attribute__((address_space(3))) v4i* lptr_v4i;

__device__ __forceinline__ void async_copy_f4_to_lds(const float4* gsrc, float4* ldst) {
#if __has_builtin(__builtin_amdgcn_global_load_async_to_lds_b128)
  __builtin_amdgcn_global_load_async_to_lds_b128(
      (gptr_v4i)(uintptr_t)gsrc,
      (lptr_v4i)(unsigned)(uintptr_t)ldst,
      /*offset=*/0, /*cpol=*/0);
#else
  unsigned            l = (unsigned)(uintptr_t)ldst;
  unsigned long long  g = (unsigned long long)(uintptr_t)gsrc;
  asm volatile("global_load_async_to_lds_b128 %0, %1, off"
               :: "v"(l), "v"(g)
               : "memory");
#endif
}

__device__ __forceinline__ void wait_async0() {
#if __has_builtin(__builtin_amdgcn_s_wait_asynccnt)
  __builtin_amdgcn_s_wait_asynccnt(0);
#else
  asm volatile("s_wait_asynccnt 0x0" ::: "memory");
#endif
}

__device__ __forceinline__ float fast_rsq(float x) {
#if __has_builtin(__builtin_amdgcn_rsqf)
  return __builtin_amdgcn_rsqf(x);      // v_rsq_f32
#else
  return rsqrtf(x);
#endif
}

__device__ __forceinline__ float fast_exp2(float x) {
#if __has_builtin(__builtin_amdgcn_exp2f)
  return __builtin_amdgcn_exp2f(x);     // v_exp_f32
#else
  return exp2f(x);
#endif
}

#define KMORSE 2.8853900817779268f      // log2(e) / w, w = 0.5

__global__ void pack_particles(const float* __restrict__ pos,
                               const float* __restrict__ rad,
                               const int*   __restrict__ act,
                               float4*      __restrict__ packed,
                               int n) {
  int i = blockIdx.x * blockDim.x + threadIdx.x;
  if (i < n) {
    float4 p;
    p.x = pos[3 * i + 0];
    p.y = pos[3 * i + 1];
    p.z = pos[3 * i + 2];
    p.w = act[i] ? fmaf(KMORSE, rad[i], 0.5f) : -1.0e9f;
    packed[i] = p;
  }
}

// One 256-particle j-tile. DIAG: tile contains the i==j diagonal (1 of 16
// tiles) -> mask it (also kills the 0*inf NaN at r=0). Non-DIAG tiles skip
// the compare+cndmask entirely.
template <bool DIAG>
__device__ __forceinline__ void tile_pass(const float4* __restrict__ tp,
                                          const float4 pi, const int i,
                                          const int jTileBase,
                                          float& fx, float& fy, float& fz) {
#pragma unroll 8
  for (int jj = 0; jj < TILE; ++jj) {
    const float4 pj = tp[jj];                   // ds_load_b128, broadcast
    const float dx = pi.x - pj.x;
    const float dy = pi.y - pj.y;
    const float dz = pi.z - pj.z;
    const float d2   = fmaf(dx, dx, fmaf(dy, dy, dz * dz));
    const float rinv = fast_rsq(d2);
    const float r    = d2 * rinv;               // sqrt(d2)
    const float E    = fast_exp2(fmaf(-KMORSE, r, pi.w + pj.w));  // E = 2e
    float s = fmaf(E, E, -(E + E)) * rinv;      // 4(e^2 - e)/r
    if (DIAG) s = (jTileBase + jj != i) ? s : 0.0f;
    fx = fmaf(s, dx, fx);
    fy = fmaf(s, dy, fy);
    fz = fmaf(s, dz, fz);
  }
}

__global__ __launch_bounds__(TILE)
void morse_forces(const float4* __restrict__ packed,
                  float*        __restrict__ out,
                  int n) {
  __shared__ float4 tile[2][TILE];

  const int tid      = threadIdx.x;
  const int iBase    = blockIdx.x * TILE;
  const int i        = iBase + tid;
  const int jBase    = blockIdx.y * JCHUNK;
  const int numTiles = JCHUNK / TILE;

  const float4 pi = packed[i];
  float fx = 0.0f, fy = 0.0f, fz = 0.0f;

  async_copy_f4_to_lds(&packed[jBase + tid], &tile[0][tid]);
  wait_async0();
  __syncthreads();

  int buf = 0;
  for (int t = 0; t < numTiles; ++t) {
    if (t + 1 < numTiles)
      async_copy_f4_to_lds(&packed[jBase + (t + 1) * TILE + tid],
                           &tile[buf ^ 1][tid]);

    const int jTileBase = jBase + t * TILE;
    if (jTileBase == iBase)       // only this aligned tile holds the diagonal
      tile_pass<true >(tile[buf], pi, i, jTileBase, fx, fy, fz);
    else
      tile_pass<false>(tile[buf], pi, i, jTileBase, fx, fy, fz);

    wait_async0();
    __syncthreads();
    buf ^= 1;
  }

  if (i < n) {
    unsafeAtomicAdd(&out[3 * i + 0], fx);   // global_atomic_add_f32
    unsafeAtomicAdd(&out[3 * i + 1], fy);
    unsafeAtomicAdd(&out[3 * i + 2], fz);
  }
}

extern "C" void kernel_launch(void* const* d_in, const int* in_sizes, int n_in,
                              void* d_out, int out_size, void* d_ws, size_t ws_size,
                              hipStream_t stream) {
  (void)n_in; (void)ws_size;
  const float* pos = (const float*)d_in[0];   // [N,3] f32
  const float* rad = (const float*)d_in[1];   // [N]   f32
  const int*   act = (const int*)d_in[2];     // [N]   integer mask
  float*       out = (float*)d_out;           // [N,3] f32
  const int n = in_sizes[1];                  // N = 4096
  float4* packed = (float4*)d_ws;             // N * 16 bytes scratch

  (void)hipMemsetAsync(d_out, 0, (size_t)out_size * sizeof(float), stream);

  pack_particles<<<(n + 255) / 256, 256, 0, stream>>>(pos, rad, act, packed, n);

  dim3 grid(n / TILE, n / JCHUNK);            // (16, 8) blocks of 256 (8 wave32)
  morse_forces<<<grid, TILE, 0, stream>>>(packed, out, n);
}